// AttnFusion_88673894793860
// MI455X (gfx1250) — compile-verified
//
#include <hip/hip_runtime.h>
#include <math.h>

typedef float v2f __attribute__((ext_vector_type(2)));
typedef float v8f __attribute__((ext_vector_type(8)));

#define EPS_F 1e-6f
#define LN2_F 0.69314718055994530942f
#define L2E_F 1.44269504088896340736f

// ---- raw gfx1250 transcendental paths (branchless), guarded fallbacks ----
__device__ __forceinline__ float fast_log2(float x) {
#if __has_builtin(__builtin_amdgcn_logf)
    return __builtin_amdgcn_logf(x);            // v_log_f32
#else
    return __log2f(x);
#endif
}
__device__ __forceinline__ float fast_exp2(float x) {
#if __has_builtin(__builtin_amdgcn_exp2f)
    return __builtin_amdgcn_exp2f(x);           // v_exp_f32
#else
    return exp2f(x);
#endif
}
__device__ __forceinline__ float fast_rcp(float x) {
#if __has_builtin(__builtin_amdgcn_rcpf)
    return __builtin_amdgcn_rcpf(x);            // v_rcp_f32
#else
    return 1.0f / x;
#endif
}
__device__ __forceinline__ float fast_tanh(float x) {
#if __has_builtin(__builtin_amdgcn_tanhf)
    return __builtin_amdgcn_tanhf(x);           // v_tanh_f32 (gfx1250 trans op)
#else
    const float e = fast_exp2(x * (2.0f * L2E_F));
    return 1.0f - 2.0f * fast_rcp(e + 1.0f);    // branchless fallback
#endif
}
__device__ __forceinline__ float clamp_eps(float x) {
#if __has_builtin(__builtin_amdgcn_fmed3f)
    return __builtin_amdgcn_fmed3f(x, EPS_F, 1.0f - EPS_F);  // v_med3_f32
#else
    return fminf(fmaxf(x, EPS_F), 1.0f - EPS_F);
#endif
}
// ds_swizzle group-of-32, SWAPX16: xor=0x10, or=0, and=0x1f -> 0x401F
__device__ __forceinline__ float swap16(float x) {
    return __int_as_float(__builtin_amdgcn_ds_swizzle(__float_as_int(x), 0x401F));
}

// One 32-row tile, log2-domain: u=log2(p/(1-p)); h=tanh((ln2*W1)u + b1);
// s=W2 h + b2; a=softmax(s); prob = 1/(1+2^(-sum a*u))  [ln2*log2e == 1].
template <bool GUARD>
__device__ __forceinline__ void process32(
    int row, int n, bool lo, v2f A,     // A is ln2-prescaled W1 (duplicated rows)
    const float* __restrict__ p,
    const float* __restrict__ b1,
    const float* __restrict__ W2,
    const float* __restrict__ b2,
    float* __restrict__ out_prob,
    float* __restrict__ out_a)
{
    int lrow = row;
    if (GUARD) lrow = (row < n) ? row : (n - 1);   // clamp: EXEC stays all-1s
    const float* pp = p + (size_t)lrow * 3;
    const float p0 = clamp_eps(__builtin_nontemporal_load(pp + 0));
    const float p1 = clamp_eps(__builtin_nontemporal_load(pp + 1));
    const float p2 = clamp_eps(__builtin_nontemporal_load(pp + 2));

    // u = log2(p) - log2(1-p)   (true logit z = ln2 * u; ln2 folded into A)
    const float u0 = fast_log2(p0) - fast_log2(1.0f - p0);
    const float u1 = fast_log2(p1) - fast_log2(1.0f - p1);
    const float u2 = fast_log2(p2) - fast_log2(1.0f - p2);

    // B = U^T tile (4x16 f32 layout): V0 = {K=0 | K=2}, V1 = {K=1 | K=3(pad 0)}
    const float t0 = swap16(u0);
    const float t1 = swap16(u1);
    const float t2 = swap16(u2);
    v2f B0, B1;
    B0.x = lo ? u0 : t2;  B0.y = lo ? u1 : 0.0f;   // batch rows base+0..15
    B1.x = lo ? t0 : u2;  B1.y = lo ? t1 : 0.0f;   // batch rows base+16..31

    v8f C0 = {0.f,0.f,0.f,0.f,0.f,0.f,0.f,0.f};
    v8f C1 = {0.f,0.f,0.f,0.f,0.f,0.f,0.f,0.f};
    // D = (ln2*W1) x U^T : lane n holds h_pre[row n][hid 0..7] across 8 dwords
    C0 = __builtin_amdgcn_wmma_f32_16x16x4_f32(false, A, false, B0, (short)0, C0, false, false);
    C1 = __builtin_amdgcn_wmma_f32_16x16x4_f32(false, A, false, B1, (short)0, C1, false, false);

    float h[8];
#pragma unroll
    for (int r = 0; r < 8; ++r)
        h[r] = fast_tanh((lo ? C0[r] : C1[r]) + b1[r]);

    // 8 -> 3 linear; weights uniform -> SGPR operands
    float s0 = b2[0], s1 = b2[1], s2 = b2[2];
#pragma unroll
    for (int r = 0; r < 8; ++r) {
        s0 = fmaf(h[r], W2[r],      s0);
        s1 = fmaf(h[r], W2[8 + r],  s1);
        s2 = fmaf(h[r], W2[16 + r], s2);
    }

    // softmax over 3 (|s| <= ~3.2: tanh in (-1,1), |W2| <= 1/sqrt(8); no max guard)
    const float e0 = fast_exp2(s0 * L2E_F);
    const float e1 = fast_exp2(s1 * L2E_F);
    const float e2 = fast_exp2(s2 * L2E_F);
    const float inv = fast_rcp(e0 + e1 + e2);
    const float a0 = e0 * inv, a1 = e1 * inv, a2 = e2 * inv;

    // fused sigmoid: sigmoid(ln2 * udot) = 1/(1 + 2^-udot)
    const float udot = fmaf(a2, u2, fmaf(a1, u1, a0 * u0));
    const float fp = fast_rcp(1.0f + fast_exp2(-udot));

    if (!GUARD || row < n) {
        __builtin_nontemporal_store(fp, out_prob + row);
        float* oa = out_a + (size_t)row * 3;
        __builtin_nontemporal_store(a0, oa + 0);
        __builtin_nontemporal_store(a1, oa + 1);
        __builtin_nontemporal_store(a2, oa + 2);
    }
}

constexpr int ITER = 4;                       // points per thread
constexpr int BLOCK = 256;                    // 8 wave32s
constexpr int ROWS_PER_BLOCK = BLOCK * ITER;  // 1024

__global__ __launch_bounds__(BLOCK) void attn_fusion_wmma_kernel(
    const float* __restrict__ p,    // [n,3]
    const float* __restrict__ W1,   // [8,3]
    const float* __restrict__ b1,   // [8]
    const float* __restrict__ W2,   // [3,8]
    const float* __restrict__ b2,   // [3]
    float* __restrict__ out_prob,   // [n]
    float* __restrict__ out_a,      // [n,3]
    int n)
{
    const int lane = threadIdx.x & 31;
    const int wave = threadIdx.x >> 5;
    const bool lo  = lane < 16;

    // A(16x4) = ln2 * W1, rows duplicated into M=0..7 and M=8..15:
    // V0 = {lanes0-15: (M,K=0) | lanes16-31: (M,K=2)}, V1 = {K=1 | K=3(pad 0)}
    const int m8 = lane & 7;
    v2f A;
    A.x = LN2_F * W1[3 * m8 + (lo ? 0 : 2)];
    A.y = lo ? (LN2_F * W1[3 * m8 + 1]) : 0.0f;

    const int base = blockIdx.x * ROWS_PER_BLOCK + wave * (32 * ITER) + lane;

    if ((size_t)(blockIdx.x + 1) * ROWS_PER_BLOCK <= (size_t)n) {
        // hot path: whole block in range, zero per-lane guards (uniform branch)
#pragma unroll
        for (int it = 0; it < ITER; ++it)
            process32<false>(base + it * 32, n, lo, A, p, b1, W2, b2, out_prob, out_a);
    } else {
#pragma unroll
        for (int it = 0; it < ITER; ++it)
            process32<true>(base + it * 32, n, lo, A, p, b1, W2, b2, out_prob, out_a);
    }
}

extern "C" void kernel_launch(void* const* d_in, const int* in_sizes, int n_in,
                              void* d_out, int out_size, void* d_ws, size_t ws_size,
                              hipStream_t stream) {
    (void)n_in; (void)out_size; (void)d_ws; (void)ws_size;
    const float* p  = (const float*)d_in[0];
    const float* W1 = (const float*)d_in[1];
    const float* b1 = (const float*)d_in[2];
    const float* W2 = (const float*)d_in[3];
    const float* b2 = (const float*)d_in[4];
    const int n = in_sizes[0] / 3;

    float* out_prob = (float*)d_out;       // [n]
    float* out_a    = (float*)d_out + n;   // [n,3]

    const int blocks = (n + ROWS_PER_BLOCK - 1) / ROWS_PER_BLOCK;
    attn_fusion_wmma_kernel<<<blocks, BLOCK, 0, stream>>>(
        p, W1, b1, W2, b2, out_prob, out_a, n);
}